// FarthestPointSample_15144054685986
// MI455X (gfx1250) — compile-verified
//
#include <hip/hip_runtime.h>
#include <cstdint>

// ---------------------------------------------------------------------------
// Farthest Point Sampling (B=16, N=32768, C=128, S=1024) for MI455X (gfx1250)
//
// Kernel 1 (fps_select): one workgroup (1024 thr = 32 wave32) per batch.
//   - min_d register-resident (32 VGPRs/thread) -> zero HBM traffic for it.
//   - coords re-read from L2 each iteration (p is 6MB, L2 is 192MB).
//   - argmax: wave32 ds_swizzle butterfly + LDS cross-wave stage.
//   - one global_prefetch_b8 per 128B cacheline to warm L2 at start.
// Kernel 2 (fps_gather): TDM tensor_load_to_lds stages the 1024 selected
//   indices into LDS (s_wait_tensorcnt), then coalesced gather of x.
// ---------------------------------------------------------------------------

#define FPS_B    16
#define FPS_N    32768
#define FPS_C    128
#define FPS_S    1024
#define FPS_TPB  1024
#define FPS_PPT  (FPS_N / FPS_TPB)    // 32 points per thread
#define FPS_CSPLIT 8
#define FPS_CPB  (FPS_C / FPS_CSPLIT) // 16 channels per gather block

typedef unsigned int u32x4 __attribute__((ext_vector_type(4)));
typedef int          i32x4 __attribute__((ext_vector_type(4)));
typedef int          i32x8 __attribute__((ext_vector_type(8)));

__device__ __forceinline__ void argmax_combine(float& v, int& i, float ov, int oi) {
  // prefer larger value; on tie prefer smaller index (argmax-first semantics)
  if (ov > v || (ov == v && oi < i)) { v = ov; i = oi; }
}

// wave32 butterfly argmax via ds_swizzle (group-of-32 XOR mode:
// offset = xor_mask<<10 | and_mask(0x1f))
#define FPS_SWZ_STEP(MASK)                                                    \
  {                                                                           \
    float ov_ = __int_as_float(__builtin_amdgcn_ds_swizzle(                   \
        __float_as_int(v), ((MASK) << 10) | 0x1f));                           \
    int oi_ = __builtin_amdgcn_ds_swizzle(i, ((MASK) << 10) | 0x1f);          \
    argmax_combine(v, i, ov_, oi_);                                           \
  }

__device__ __forceinline__ void wave_argmax(float& v, int& i) {
  FPS_SWZ_STEP(16)
  FPS_SWZ_STEP(8)
  FPS_SWZ_STEP(4)
  FPS_SWZ_STEP(2)
  FPS_SWZ_STEP(1)
}

__global__ __launch_bounds__(FPS_TPB, 1) void fps_select_kernel(
    const float* __restrict__ p, float* __restrict__ p_out,
    int* __restrict__ idx_out) {
  const int b    = blockIdx.x;
  const int t    = threadIdx.x;
  const int wave = t >> 5;
  const int lane = t & 31;
  const float* __restrict__ pb = p + (size_t)b * FPS_N * 3;

  __shared__ float s_val[FPS_TPB / 32];
  __shared__ int   s_idx[FPS_TPB / 32];
  __shared__ float s_q[3];

  // Warm L2 with this batch's coords: one prefetch per 128B cacheline.
  // 384KB / 128B = 3072 lines -> 3 prefetches per thread (1024 threads).
  {
    const char* pc = (const char*)pb;
#pragma unroll
    for (int k = 0; k < 3; ++k)
      __builtin_prefetch(pc + (size_t)(t + k * FPS_TPB) * 128, 0, 1);
  }

  float mind[FPS_PPT];
#pragma unroll
  for (int k = 0; k < FPS_PPT; ++k) mind[k] = 3.402823466e38f;

  // First selected index is 0 (matches reference).
  float qx = pb[0], qy = pb[1], qz = pb[2];
  if (t == 0) {
    float* po = p_out + (size_t)b * FPS_S * 3;
    po[0] = qx; po[1] = qy; po[2] = qz;
    idx_out[b * FPS_S] = 0;
  }

  for (int s = 1; s < FPS_S; ++s) {
    float bv = -1.0f;
    int   bi = 0;
#pragma unroll
    for (int k = 0; k < FPS_PPT; ++k) {
      const int n = t + k * FPS_TPB;            // interleaved -> coalesced b96
      const float* pp = pb + (size_t)n * 3;
      float dx = pp[0] - qx;
      float dy = pp[1] - qy;
      float dz = pp[2] - qz;
      float d  = fmaf(dz, dz, fmaf(dy, dy, dx * dx));
      float m  = fminf(mind[k], d);
      mind[k]  = m;
      if (m > bv) { bv = m; bi = n; }           // strict > keeps smallest n
    }
    wave_argmax(bv, bi);
    if (lane == 0) { s_val[wave] = bv; s_idx[wave] = bi; }
    __syncthreads();
    if (wave == 0) {
      float v = s_val[lane];
      int   i = s_idx[lane];
      wave_argmax(v, i);
      if (lane == 0) {
        const float* wp = pb + (size_t)i * 3;   // L2/WGP$ hit
        float wx = wp[0], wy = wp[1], wz = wp[2];
        s_q[0] = wx; s_q[1] = wy; s_q[2] = wz;
        float* po = p_out + ((size_t)b * FPS_S + (size_t)s) * 3;
        po[0] = wx; po[1] = wy; po[2] = wz;
        idx_out[b * FPS_S + s] = i;
      }
    }
    __syncthreads();
    qx = s_q[0]; qy = s_q[1]; qz = s_q[2];
  }
}

__global__ __launch_bounds__(FPS_S, 1) void fps_gather_kernel(
    const float* __restrict__ x, const int* __restrict__ idx_in,
    float* __restrict__ x_out) {
  const int b  = blockIdx.x;
  const int c0 = blockIdx.y * FPS_CPB;
  const int t  = threadIdx.x;

  __shared__ int s_sel[FPS_S];  // 4KB of selected indices for this batch

  if (t < 32) {  // one wave issues the TDM descriptor (EXEC ignored by TDM)
    const uint64_t ga  = (uint64_t)(uintptr_t)(idx_in + (size_t)b * FPS_S);
    const uint32_t lds = (uint32_t)(uintptr_t)(&s_sel[0]);

    // D# group 0: count=1 | lds_addr | 57-bit global addr | type=2
    u32x4 g0;
    g0[0] = 1u;
    g0[1] = lds;
    g0[2] = (uint32_t)ga;
    g0[3] = ((uint32_t)(ga >> 32) & 0x01FFFFFFu) | 0x80000000u;

    // D# group 1: data_size=4B; tensor_dim0=1024, tensor_dim1=1,
    // tile_dim0=1024, tile_dim1=1, tensor_dim0_stride=1024
    i32x8 g1;
    g1[0] = 2 << 16;                 // data_size = 2 (4 bytes), mask/flags = 0
    g1[1] = (int)(FPS_S << 16);      // tensor_dim0[15:0] in [63:48]
    g1[2] = 1 << 16;                 // tensor_dim0 hi = 0, tensor_dim1 = 1
    g1[3] = (int)(FPS_S << 16);      // tensor_dim1 hi = 0, tile_dim0 = 1024
    g1[4] = 1;                       // tile_dim1 = 1, tile_dim2 = 0
    g1[5] = FPS_S;                   // tensor_dim0_stride lo
    g1[6] = 0;                       // stride hi / tensor_dim1_stride lo
    g1[7] = 0;

    i32x4 g2 = {0, 0, 0, 0};
    i32x4 g3 = {0, 0, 0, 0};
#if defined(__clang_major__) && (__clang_major__ >= 23)
    i32x8 g4 = {0, 0, 0, 0, 0, 0, 0, 0};
    __builtin_amdgcn_tensor_load_to_lds(g0, g1, g2, g3, g4, 0);
#else
    __builtin_amdgcn_tensor_load_to_lds(g0, g1, g2, g3, 0);
#endif
    __builtin_amdgcn_s_wait_tensorcnt(0);
  }
  __syncthreads();

  const int off = s_sel[t];
  const float* __restrict__ xb = x     + ((size_t)b * FPS_C + (size_t)c0) * FPS_N;
  float*       __restrict__ ob = x_out + ((size_t)b * FPS_C + (size_t)c0) * FPS_S;
#pragma unroll
  for (int c = 0; c < FPS_CPB; ++c)
    ob[(size_t)c * FPS_S + t] = xb[(size_t)c * FPS_N + (size_t)off];
}

extern "C" void kernel_launch(void* const* d_in, const int* in_sizes, int n_in,
                              void* d_out, int out_size, void* d_ws,
                              size_t ws_size, hipStream_t stream) {
  (void)in_sizes; (void)n_in; (void)out_size; (void)ws_size;

  const float* p = (const float*)d_in[0];   // [B, N, 3] f32
  const float* x = (const float*)d_in[1];   // [B, C, N] f32

  float* p_out = (float*)d_out;                                   // [B,S,3]
  float* x_out = p_out + (size_t)FPS_B * FPS_S * 3;               // [B,C,S]
  int*   idx_ws = (int*)d_ws;                                     // [B,S] int32

  fps_select_kernel<<<FPS_B, FPS_TPB, 0, stream>>>(p, p_out, idx_ws);
  fps_gather_kernel<<<dim3(FPS_B, FPS_CSPLIT), FPS_S, 0, stream>>>(x, idx_ws,
                                                                   x_out);
}